// TIADGCN_22127671509734
// MI455X (gfx1250) — compile-verified
//
#include <hip/hip_runtime.h>
#include <math.h>

#define CDIV(a,b) (((a)+(b)-1)/(b))

typedef __attribute__((ext_vector_type(2))) float v2f;
typedef __attribute__((ext_vector_type(8))) float v8f;

__device__ __forceinline__ v8f wmma4(v2f a, v2f b, v8f c) {
  // V_WMMA_F32_16X16X4_F32 : D = A(16x4) * B(4x16) + C, fp32 exact
  return __builtin_amdgcn_wmma_f32_16x16x4_f32(false, a, false, b, (short)0, c, false, false);
}

// ---------------------------------------------------------------------------
// Dense channel-mix GEMM: Y[b,o,n] = act( sum_i W[o,i]*X[b,i,n] + bias[o] ) (+R)
// X,Y layout (16, C, Nn). W row-major (M x Kd). Wave computes 16(M) x 64(N).
// Out-of-range rows/cols use clamped addresses (unconditional loads); their
// garbage results are simply not stored.
// ---------------------------------------------------------------------------
__global__ __launch_bounds__(256) void k_dense(
    const float* __restrict__ W, const float* __restrict__ bias,
    const float* __restrict__ X, const float* __restrict__ R,
    float* __restrict__ Y, int M, int Kd, int Nn, int act)
{
  const long Ncols  = 16L * Nn;
  const int tilesM = (M + 15) >> 4;
  const int tilesG = (int)((Ncols + 63) >> 6);
  int wid = blockIdx.x * 8 + (threadIdx.x >> 5);
  int tm = wid % tilesM, tg = wid / tilesM;
  if (tg >= tilesG) return;
  int lane = threadIdx.x & 31, half = lane >> 4, lr = lane & 15, kadd = half << 1;

  int m = tm * 16 + lr;
  int mc = m < M ? m : M - 1;
  long xb[4];
#pragma unroll
  for (int s = 0; s < 4; ++s) {
    long j = (long)tg * 64 + s * 16 + lr;
    long jc = j < Ncols ? j : Ncols - 1;
    int b = (int)(jc / Nn); int n = (int)(jc - (long)b * Nn);
    xb[s] = (long)b * Kd * Nn + n;
  }

  v8f acc[4] = {};
  for (int k0 = 0; k0 < Kd; k0 += 4) {
    int kk = k0 + kadd;
    v2f a;
    a.x = W[(long)mc * Kd + kk];
    a.y = W[(long)mc * Kd + kk + 1];
#pragma unroll
    for (int s = 0; s < 4; ++s) {
      v2f bv;
      bv.x = X[xb[s] + (long)kk * Nn];
      bv.y = X[xb[s] + (long)(kk + 1) * Nn];
      acc[s] = wmma4(a, bv, acc[s]);
    }
  }

#pragma unroll
  for (int s = 0; s < 4; ++s) {
    long j = (long)tg * 64 + s * 16 + lr;
    if (j >= Ncols) continue;
    int b = (int)(j / Nn); int n = (int)(j - (long)b * Nn);
#pragma unroll
    for (int g = 0; g < 8; ++g) {
      int o = tm * 16 + g + (half << 3);
      if (o < M) {
        float v = acc[s][g] + bias[o];
        if (act == 1) v = fmaxf(v, 0.f);
        long oi = ((long)b * M + o) * Nn + n;
        if (R) v += R[oi];
        Y[oi] = v;
      }
    }
  }
}

// ---------------------------------------------------------------------------
// Graph GEMM: P[b,n,m] = relu(scale * sum_c Q[b,c,n] * Bmat[c,m])
// Bmat = memory (128 x Nn) or Q_b (gram). Wave computes 16 x 64.
// ---------------------------------------------------------------------------
__global__ __launch_bounds__(256) void k_graphgemm(
    const float* __restrict__ Q, const float* __restrict__ Bsrc,
    float* __restrict__ P, int Nn, int useQ, float scale)
{
  int b = blockIdx.y;
  const int tilesM = (Nn + 15) >> 4;   // 56
  const int tilesG = (Nn + 63) >> 6;   // 14
  int wid = blockIdx.x * 8 + (threadIdx.x >> 5);
  int tm = wid % tilesM, tg = wid / tilesM;
  if (tg >= tilesG) return;
  int lane = threadIdx.x & 31, half = lane >> 4, lr = lane & 15, kadd = half << 1;
  const float* Qb = Q + (long)b * 128 * Nn;
  const float* Bb = useQ ? Qb : Bsrc;
  int nrow = tm * 16 + lr;
  int nr = nrow < Nn ? nrow : Nn - 1;
  int mc[4];
#pragma unroll
  for (int s = 0; s < 4; ++s) { int mm = tg * 64 + s * 16 + lr; mc[s] = mm < Nn ? mm : Nn - 1; }

  v8f acc[4] = {};
  for (int k0 = 0; k0 < 128; k0 += 4) {
    int kk = k0 + kadd;
    v2f a;
    a.x = Qb[(long)kk * Nn + nr];
    a.y = Qb[(long)(kk + 1) * Nn + nr];
#pragma unroll
    for (int s = 0; s < 4; ++s) {
      v2f bv;
      bv.x = Bb[(long)kk * Nn + mc[s]];
      bv.y = Bb[(long)(kk + 1) * Nn + mc[s]];
      acc[s] = wmma4(a, bv, acc[s]);
    }
  }

#pragma unroll
  for (int s = 0; s < 4; ++s) {
    int mcol = tg * 64 + s * 16 + lr;
    if (mcol >= Nn) continue;
#pragma unroll
    for (int g = 0; g < 8; ++g) {
      int r = tm * 16 + g + (half << 3);
      if (r < Nn) P[((long)b * Nn + r) * Nn + mcol] = fmaxf(acc[s][g] * scale, 0.f);
    }
  }
}

// ---------------------------------------------------------------------------
// Aggregation GEMM: Agg[b,c,n] = sum_m V[b,c,m] * Adj[b,n,m]  (B read transposed)
// K = Nn = 883 -> main loop to 880 + masked tail (value selects, no branches).
// ---------------------------------------------------------------------------
__global__ __launch_bounds__(256) void k_agg(
    const float* __restrict__ V, const float* __restrict__ Adj,
    float* __restrict__ Agg, int Nn)
{
  int b = blockIdx.y;
  const int tilesG = (Nn + 63) >> 6;   // 14
  int wid = blockIdx.x * 8 + (threadIdx.x >> 5);
  int tm = wid % 8, tg = wid / 8;      // tilesM = 8 (C = 128)
  if (tg >= tilesG) return;
  int lane = threadIdx.x & 31, half = lane >> 4, lr = lane & 15, kadd = half << 1;
  const float* Vb = V + (long)b * 128 * Nn;
  const float* Ab = Adj + (long)b * Nn * Nn;
  int crow = tm * 16 + lr;             // < 128 always
  int nc[4];
#pragma unroll
  for (int s = 0; s < 4; ++s) { int n = tg * 64 + s * 16 + lr; nc[s] = n < Nn ? n : Nn - 1; }

  v8f acc[4] = {};
  const int Kmain = Nn & ~3;           // 880
  for (int k0 = 0; k0 < Kmain; k0 += 4) {
    int kk = k0 + kadd;
    v2f a;
    a.x = Vb[(long)crow * Nn + kk];
    a.y = Vb[(long)crow * Nn + kk + 1];
#pragma unroll
    for (int s = 0; s < 4; ++s) {
      v2f bv;
      bv.x = Ab[(long)nc[s] * Nn + kk];
      bv.y = Ab[(long)nc[s] * Nn + kk + 1];
      acc[s] = wmma4(a, bv, acc[s]);
    }
  }
  { // tail: kk in {880, 882}; mask out-of-range K elements by value
    int kk = Kmain + kadd;
    float m0 = (kk     < Nn) ? 1.f : 0.f;
    float m1 = (kk + 1 < Nn) ? 1.f : 0.f;
    int e0 = kk     < Nn ? kk     : Nn - 1;
    int e1 = kk + 1 < Nn ? kk + 1 : Nn - 1;
    v2f a;
    a.x = Vb[(long)crow * Nn + e0] * m0;
    a.y = Vb[(long)crow * Nn + e1] * m1;
#pragma unroll
    for (int s = 0; s < 4; ++s) {
      v2f bv;
      bv.x = Ab[(long)nc[s] * Nn + e0] * m0;
      bv.y = Ab[(long)nc[s] * Nn + e1] * m1;
      acc[s] = wmma4(a, bv, acc[s]);
    }
  }

#pragma unroll
  for (int s = 0; s < 4; ++s) {
    int ncol = tg * 64 + s * 16 + lr;
    if (ncol >= Nn) continue;
#pragma unroll
    for (int g = 0; g < 8; ++g) {
      int c = tm * 16 + g + (half << 3);
      Agg[((long)b * 128 + c) * Nn + ncol] = acc[s][g];
    }
  }
}

// ---------------------------------------------------------------------------
// Temporal conv as im2col-GEMM, templated on kernel width / pad mode.
// src (16,64,Nn,Lsrc); logical input z[ts] = src[..., toff + tstride*ts];
// left pad padL (EDGE: clamp, else zero via value select). Wave = 16 x 32.
// ---------------------------------------------------------------------------
template<int KW, int EDGE>
__global__ __launch_bounds__(256) void k_convgemm(
    const float* __restrict__ W, const float* __restrict__ bias,
    const float* __restrict__ src, float* __restrict__ dst,
    int Nn, int Lsrc, int Lout, int padL, int tstride, int toff, int act)
{
  const int Kd = 64 * KW;
  const long Ncols = 16L * Nn * Lout;
  const int tilesG = (int)((Ncols + 31) >> 5);
  int wid = blockIdx.x * 8 + (threadIdx.x >> 5);
  int tm = wid & 3, tg = wid >> 2;     // tilesM = 4 (M = 64)
  if (tg >= tilesG) return;
  int lane = threadIdx.x & 31, half = lane >> 4, lr = lane & 15, kadd = half << 1;
  int m = tm * 16 + lr;                // < 64 always
  long istride = (long)Nn * Lsrc;

  long sb[2]; int tt[2];
#pragma unroll
  for (int s = 0; s < 2; ++s) {
    long j = (long)tg * 32 + s * 16 + lr;
    long jc = j < Ncols ? j : Ncols - 1;
    int b = (int)(jc / ((long)Nn * Lout));
    long r = jc - (long)b * Nn * Lout;
    int n = (int)(r / Lout);
    tt[s] = (int)(r - (long)n * Lout);
    sb[s] = ((long)b * 64 * Nn + n) * Lsrc + toff;
  }

  v8f acc[2] = {};
  for (int k0 = 0; k0 < Kd; k0 += 4) {
    int kk = k0 + kadd;
    int i0 = kk / KW,       kw0 = kk - i0 * KW;
    int i1 = (kk + 1) / KW, kw1 = (kk + 1) - i1 * KW;
    v2f a;
    a.x = W[(long)m * Kd + kk];
    a.y = W[(long)m * Kd + kk + 1];
#pragma unroll
    for (int s = 0; s < 2; ++s) {
      int ts0 = tt[s] + kw0 - padL; int t0 = ts0 < 0 ? 0 : ts0;
      int ts1 = tt[s] + kw1 - padL; int t1 = ts1 < 0 ? 0 : ts1;
      float v0 = src[sb[s] + (long)i0 * istride + (long)tstride * t0];
      float v1 = src[sb[s] + (long)i1 * istride + (long)tstride * t1];
      if (!EDGE) { v0 = ts0 < 0 ? 0.f : v0; v1 = ts1 < 0 ? 0.f : v1; }
      v2f bv; bv.x = v0; bv.y = v1;
      acc[s] = wmma4(a, bv, acc[s]);
    }
  }

#pragma unroll
  for (int s = 0; s < 2; ++s) {
    long j = (long)tg * 32 + s * 16 + lr;
    if (j >= Ncols) continue;
    int b = (int)(j / ((long)Nn * Lout));
    long r = j - (long)b * Nn * Lout;
    int n = (int)(r / Lout);
    int t = (int)(r - (long)n * Lout);
#pragma unroll
    for (int g = 0; g < 8; ++g) {
      int o = tm * 16 + g + (half << 3);
      float v = acc[s][g] + bias[o];
      v = (act == 2) ? tanhf(v) : fmaxf(v, 0.f);
      dst[(((long)b * 64 + o) * Nn + n) * Lout + t] = v;
    }
  }
}

// ---------------------------------------------------------------------------
// Elementwise kernels
// ---------------------------------------------------------------------------
__global__ void k_start(const float* __restrict__ W, const float* __restrict__ bs,
                        const float* __restrict__ hist, float* __restrict__ x,
                        int Nn, long total)
{
  long i = (long)blockIdx.x * 256 + threadIdx.x;
  if (i >= total) return;
  int t = i % 12; long r = i / 12; int n = r % Nn; r /= Nn; int c = r % 64; int b = r / 64;
  const float* h = hist + ((long)b * 3 * Nn + n) * 12 + t;
  long ch = (long)Nn * 12;
  x[i] = bs[c] + W[c*3+0]*h[0] + W[c*3+1]*h[ch] + W[c*3+2]*h[2*ch];
}

// d = xo*exp(t1); c = xe*exp(t2)   (written in-place over t1,t2 buffers)
__global__ void k_comb1(const float* __restrict__ x, float* __restrict__ t1d,
                        float* __restrict__ t2c, int Nn, long total)
{
  long i = (long)blockIdx.x * 256 + threadIdx.x;
  if (i >= total) return;
  int t = i % 6; long r = i / 6; int n = r % Nn; r /= Nn; int c = r % 64; int b = r / 64;
  long xb = (((long)b * 64 + c) * Nn + n) * 12;
  float xe = x[xb + 2*t], xo = x[xb + 2*t + 1];
  t1d[i] = xo * expf(t1d[i]);
  t2c[i] = xe * expf(t2c[i]);
}

// x[...,2t] += d + t3 ; x[...,2t+1] += c + t4
__global__ void k_comb2(float* __restrict__ x, const float* __restrict__ d,
                        const float* __restrict__ cb, const float* __restrict__ t3,
                        const float* __restrict__ t4, int Nn, long total)
{
  long i = (long)blockIdx.x * 256 + threadIdx.x;
  if (i >= total) return;
  int t = i % 6; long r = i / 6; int n = r % Nn; r /= Nn; int c = r % 64; int b = r / 64;
  long xb = (((long)b * 64 + c) * Nn + n) * 12;
  x[xb + 2*t]     += d[i]  + t3[i];
  x[xb + 2*t + 1] += cb[i] + t4[i];
}

__global__ void k_build_datast(const float* __restrict__ h3, const float* __restrict__ x,
                               const float* __restrict__ hist, const float* __restrict__ tday,
                               const float* __restrict__ tweek, float* __restrict__ X0,
                               int Nn, long total)
{
  long i = (long)blockIdx.x * 256 + threadIdx.x;
  if (i >= total) return;
  int n = i % Nn; long r = i / Nn; int c = r % 128; int b = r / 128;
  float v;
  if (c < 64) {
    v = h3[((long)b * 64 + c) * Nn + n] + x[(((long)b * 64 + c) * Nn + n) * 12 + 11];
  } else {
    int cc = c - 64;
    float day = hist[(((long)b * 3 + 1) * Nn + n) * 12 + 11];
    float wk  = hist[(((long)b * 3 + 2) * Nn + n) * 12 + 11];
    int di = (int)(day * 288.0f); di = di < 0 ? 0 : (di > 287 ? 287 : di);
    int wi = (int)wk;             wi = wi < 0 ? 0 : (wi > 6 ? 6 : wi);
    v = tday[di * 64 + cc] + tweek[wi * 64 + cc];
  }
  X0[i] = v;
}

__global__ void k_sigprod(float* __restrict__ A, const float* __restrict__ Bm, long total)
{
  long i = (long)blockIdx.x * 256 + threadIdx.x;
  if (i >= total) return;
  A[i] = A[i] * (1.0f / (1.0f + expf(-Bm[i])));
}

__global__ void k_dyt(float* __restrict__ X, const float* __restrict__ a,
                      const float* __restrict__ rr, const float* __restrict__ bb,
                      int Nn, long total)
{
  long i = (long)blockIdx.x * 256 + threadIdx.x;
  if (i >= total) return;
  int c = (int)((i / Nn) % 128);
  X[i] = rr[c] * tanhf(a[0] * X[i]) + bb[c];
}

__global__ void k_encw(float* __restrict__ X, const float* __restrict__ ew,
                       int Nn, long total)
{
  long i = (long)blockIdx.x * 256 + threadIdx.x;
  if (i >= total) return;
  int n = i % Nn; int c = (int)((i / Nn) % 128);
  X[i] = X[i] * (1.0f + ew[(long)c * Nn + n]);
}

// ---------------------------------------------------------------------------
// Fused row pipeline: softmax(P1), softmax(P2), 2->8->1 MLP, softmax, top-K mask
// One block per (b,n) row of length Nn=883 (staged in LDS). Result -> P1.
// ---------------------------------------------------------------------------
__device__ __forceinline__ float rf_rmax(float v, float* red, int tid) {
  red[tid] = v; __syncthreads();
  for (int s = 128; s > 0; s >>= 1) { if (tid < s) red[tid] = fmaxf(red[tid], red[tid+s]); __syncthreads(); }
  float r = red[0]; __syncthreads(); return r;
}
__device__ __forceinline__ float rf_rsum(float v, float* red, int tid) {
  red[tid] = v; __syncthreads();
  for (int s = 128; s > 0; s >>= 1) { if (tid < s) red[tid] += red[tid+s]; __syncthreads(); }
  float r = red[0]; __syncthreads(); return r;
}
__device__ float rf_softmax(float* s, int Nn, float* red, int tid) {
  float lm = -3.4e38f;
  for (int m = tid; m < Nn; m += 256) lm = fmaxf(lm, s[m]);
  float mx = rf_rmax(lm, red, tid);
  float ls = 0.f;
  for (int m = tid; m < Nn; m += 256) { float e = expf(s[m] - mx); s[m] = e; ls += e; }
  float inv = 1.f / rf_rsum(ls, red, tid);
  for (int m = tid; m < Nn; m += 256) s[m] *= inv;
  __syncthreads();
  return inv;                 // == max of normalized row
}

__global__ __launch_bounds__(256) void k_rowfuse(
    float* __restrict__ P1, const float* __restrict__ P2,
    const float* __restrict__ f1w, const float* __restrict__ f1b,
    const float* __restrict__ f2w, const float* __restrict__ f2b,
    int Nn, int Ksel)
{
  __shared__ float s1[896], s2[896], red[256];
  int tid = threadIdx.x;
  long base = (long)blockIdx.x * Nn;
  for (int m = tid; m < Nn; m += 256) { s1[m] = P1[base + m]; s2[m] = P2[base + m]; }
  __syncthreads();
  rf_softmax(s1, Nn, red, tid);
  rf_softmax(s2, Nn, red, tid);

  float w0[8], w1[8], bb[8], w2[8];
#pragma unroll
  for (int k = 0; k < 8; ++k) { w0[k]=f1w[2*k]; w1[k]=f1w[2*k+1]; bb[k]=f1b[k]; w2[k]=f2w[k]; }
  float b2v = f2b[0];
  for (int m = tid; m < Nn; m += 256) {
    float a1 = s1[m], a2 = s2[m], sv = b2v;
#pragma unroll
    for (int k = 0; k < 8; ++k) { float h = fmaxf(w0[k]*a1 + w1[k]*a2 + bb[k], 0.f); sv += w2[k]*h; }
    s1[m] = sv;
  }
  __syncthreads();
  float inv = rf_softmax(s1, Nn, red, tid);

  // top-K threshold by bisection on normalized values in (0, inv]
  float lo = 0.f, hi = inv * 1.0000002f + 1e-12f;
  for (int it = 0; it < 28; ++it) {
    float mid = 0.5f * (lo + hi);
    float c = 0.f;
    for (int m = tid; m < Nn; m += 256) c += (s1[m] >= mid) ? 1.f : 0.f;
    float cnt = rf_rsum(c, red, tid);
    if (cnt >= (float)Ksel) lo = mid; else hi = mid;
  }
  for (int m = tid; m < Nn; m += 256) P1[base + m] = (s1[m] >= lo) ? s1[m] : 0.f;
}

// ---------------------------------------------------------------------------
// Host launcher
// ---------------------------------------------------------------------------
extern "C" void kernel_launch(void* const* d_in, const int* in_sizes, int n_in,
                              void* d_out, int out_size, void* d_ws, size_t ws_size,
                              hipStream_t stream) {
  (void)in_sizes; (void)n_in; (void)out_size; (void)ws_size;
  const float* start_w = (const float*)d_in[0];
  const float* start_b = (const float*)d_in[1];
  const float* time_day= (const float*)d_in[2];
  const float* time_wk = (const float*)d_in[3];
  const float* memoryp = (const float*)d_in[4];
  const float* fc1_w   = (const float*)d_in[5];
  const float* fc1_b   = (const float*)d_in[6];
  const float* fc2_w   = (const float*)d_in[7];
  const float* fc2_b   = (const float*)d_in[8];
  const float* dyt1_a  = (const float*)d_in[9];
  const float* dyt1_r  = (const float*)d_in[10];
  const float* dyt1_b  = (const float*)d_in[11];
  const float* dyt2_a  = (const float*)d_in[12];
  const float* dyt2_r  = (const float*)d_in[13];
  const float* dyt2_b  = (const float*)d_in[14];
  const float* enc_w   = (const float*)d_in[15];
  const float* q_w     = (const float*)d_in[16];
  const float* q_b     = (const float*)d_in[17];
  const float* v_w     = (const float*)d_in[18];
  const float* v_b     = (const float*)d_in[19];
  const float* cc_w    = (const float*)d_in[20];
  const float* cc_b    = (const float*)d_in[21];
  const float* fcst_w  = (const float*)d_in[22];
  const float* fcst_b  = (const float*)d_in[23];
  const float* reg_w   = (const float*)d_in[24];
  const float* reg_b   = (const float*)d_in[25];
  const float* idgw[4] = {(const float*)d_in[26],(const float*)d_in[28],(const float*)d_in[30],(const float*)d_in[32]};
  const float* idgb[4] = {(const float*)d_in[27],(const float*)d_in[29],(const float*)d_in[31],(const float*)d_in[33]};
  const float* tcw[4]  = {(const float*)d_in[34],(const float*)d_in[36],(const float*)d_in[38],(const float*)d_in[40]};
  const float* tcb[4]  = {(const float*)d_in[35],(const float*)d_in[37],(const float*)d_in[39],(const float*)d_in[41]};
  const float* g1w1=(const float*)d_in[42]; const float* g1b1=(const float*)d_in[43];
  const float* g1w2=(const float*)d_in[44]; const float* g1b2=(const float*)d_in[45];
  const float* g1w3=(const float*)d_in[46]; const float* g1b3=(const float*)d_in[47];
  const float* g2w1=(const float*)d_in[48]; const float* g2b1=(const float*)d_in[49];
  const float* g2w2=(const float*)d_in[50]; const float* g2b2=(const float*)d_in[51];
  const float* g2w3=(const float*)d_in[52]; const float* g2b3=(const float*)d_in[53];
  const float* hist = (const float*)d_in[54];
  float* out = (float*)d_out;

  const int Nn = 883, Ksel = 706;
  const float scale = 0.08838834764831845f;  // 1/sqrt(128)

  // workspace layout (floats), with deliberate aliasing
  float* ws = (float*)d_ws;
  const long SZ_X  = 16L*64*Nn*12;
  const long SZ_B6 = 16L*64*Nn*6;
  float* xbuf = ws;
  float* b1 = ws + SZ_X;
  float* b2 = b1 + SZ_B6;
  float* b3 = b2 + SZ_B6;
  float* b4 = b3 + SZ_B6;
  // h-chain aliases the b1..b4 region (dead by then)
  float* h0 = b1;
  float* h1 = h0 + 16L*64*Nn*10;
  float* h2 = h1 + 16L*64*Nn*7;
  float* h3 = h2 + 16L*64*Nn*4;
  // adjacency tensors alias xbuf+bufs region (dead by then)
  float* P1 = ws;
  float* P2 = ws + 16L*Nn*Nn;
  // encoder slots
  const long XOFF = SZ_X + 4*SZ_B6;
  const long SZ_S = 16L*128*Nn;
  float* S0 = ws + XOFF;
  float* S1 = S0 + SZ_S; float* S2 = S1 + SZ_S; float* S3 = S2 + SZ_S;
  float* S4 = S3 + SZ_S; float* S5 = S4 + SZ_S;

  // ---- 1) start 1x1 conv ----
  long tot = 16L*64*Nn*12;
  k_start<<<CDIV(tot,256),256,0,stream>>>(start_w,start_b,hist,xbuf,Nn,tot);

  auto conv3 = [&](const float* W,const float* bias,const float* src,float* dst,
                   int Lsrc,int Lout,int padL,int ts,int to,int act){
    long nc = 16L*Nn*Lout; int tiles = 4*(int)CDIV(nc,32);
    k_convgemm<3,1><<<CDIV(tiles,8),256,0,stream>>>(W,bias,src,dst,Nn,Lsrc,Lout,padL,ts,to,act);
  };
  auto conv4 = [&](const float* W,const float* bias,const float* src,float* dst,
                   int Lsrc,int Lout,int padL,int act){
    long nc = 16L*Nn*Lout; int tiles = 4*(int)CDIV(nc,32);
    k_convgemm<4,0><<<CDIV(tiles,8),256,0,stream>>>(W,bias,src,dst,Nn,Lsrc,Lout,padL,1,0,act);
  };
  // ---- 2) lifting scheme (edge-pad 2, width-3, tanh) ----
  conv3(idgw[0],idgb[0],xbuf,b1, 12,6, 2, 2,0, 2);   // t1 = lift(xe)
  conv3(idgw[1],idgb[1],xbuf,b2, 12,6, 2, 2,1, 2);   // t2 = lift(xo)
  long tot6 = 16L*64*Nn*6;
  k_comb1<<<CDIV(tot6,256),256,0,stream>>>(xbuf,b1,b2,Nn,tot6);   // b1=d, b2=c
  conv3(idgw[2],idgb[2],b2,b3, 6,6, 2, 1,0, 2);      // t3 = lift(c)
  conv3(idgw[3],idgb[3],b1,b4, 6,6, 2, 1,0, 2);      // t4 = lift(d)
  k_comb2<<<CDIV(tot6,256),256,0,stream>>>(xbuf,b1,b2,b3,b4,Nn,tot6);
  // ---- 3) temporal conv chain (zero-pad 1 once, width-4, relu) ----
  conv4(tcw[0],tcb[0],xbuf,h0, 12,10, 1, 1);
  conv4(tcw[1],tcb[1],h0,  h1, 10, 7, 0, 1);
  conv4(tcw[2],tcb[2],h1,  h2,  7, 4, 0, 1);
  conv4(tcw[3],tcb[3],h2,  h3,  4, 1, 0, 1);
  // ---- 4) build data_st = [h3 + x[...,-1] ; time embeddings] ----
  long totS = 16L*128*Nn;
  k_build_datast<<<CDIV(totS,256),256,0,stream>>>(h3,xbuf,hist,time_day,time_wk,S0,Nn,totS);

  auto dense = [&](const float* W,const float* bias,const float* X,const float* R,
                   float* Y,int M,int act){
    int tiles = ((M+15)/16) * (int)CDIV(16L*Nn,64);
    k_dense<<<CDIV(tiles,8),256,0,stream>>>(W,bias,X,R,Y,M,128,Nn,act);
  };
  // ---- 5) encoder: glu1 + residual ----
  dense(g1w1,g1b1,S0,nullptr,S1,128,0);
  dense(g1w2,g1b2,S0,nullptr,S2,128,0);
  k_sigprod<<<CDIV(totS,256),256,0,stream>>>(S1,S2,totS);
  dense(g1w3,g1b3,S1,S0,S2,128,0);                      // X1 -> S2
  // dyt1 (in place) -> X2 in S2
  k_dyt<<<CDIV(totS,256),256,0,stream>>>(S2,dyt1_a,dyt1_r,dyt1_b,Nn,totS);
  // ---- 6) dgcn ----
  dense(q_w,q_b,S2,nullptr,S1,128,1);                   // Q -> S1
  dense(v_w,v_b,S2,nullptr,S3,128,1);                   // V -> S3
  {
    int tiles = ((Nn+15)/16) * ((Nn+63)/64);            // 56 * 14
    dim3 g(CDIV(tiles,8), 16);
    k_graphgemm<<<g,256,0,stream>>>(S1,memoryp,P1,Nn,0,scale);  // adj1 pre-softmax
    k_graphgemm<<<g,256,0,stream>>>(S1,nullptr,P2,Nn,1,scale);  // adj2 (gram)
  }
  k_rowfuse<<<16*Nn,256,0,stream>>>(P1,P2,fc1_w,fc1_b,fc2_w,fc2_b,Nn,Ksel);
  {
    int tiles = 8 * ((Nn+63)/64);                       // 8 * 14
    dim3 g(CDIV(tiles,8), 16);
    k_agg<<<g,256,0,stream>>>(S3,P1,S4,Nn);             // Agg -> S4
  }
  dense(cc_w,cc_b,S4,S2,S5,128,1);                      // X3 = X2 + relu(cc) -> S5
  // ---- 7) glu2 + residual ----
  dense(g2w1,g2b1,S5,nullptr,S1,128,0);
  dense(g2w2,g2b2,S5,nullptr,S3,128,0);
  k_sigprod<<<CDIV(totS,256),256,0,stream>>>(S1,S3,totS);
  dense(g2w3,g2b3,S1,S5,S4,128,0);                      // X4 -> S4
  // ---- 8) enc_weight residual scale, dyt2 ----
  k_encw<<<CDIV(totS,256),256,0,stream>>>(S4,enc_w,Nn,totS);
  k_dyt<<<CDIV(totS,256),256,0,stream>>>(S4,dyt2_a,dyt2_r,dyt2_b,Nn,totS);
  // ---- 9) + fcst(X0), then regression head ----
  dense(fcst_w,fcst_b,S0,S4,S1,128,0);                  // final features -> S1
  dense(reg_w,reg_b,S1,nullptr,out,12,0);               // pred (B,12,N)
}